// ManifoldConstrainedHyperConnections_50620484551319
// MI455X (gfx1250) — compile-verified
//
#include <hip/hip_runtime.h>
#include <hip/hip_bf16.h>
#include <math.h>

#define KD 4
#define BD 2
#define SD 1024
#define DD 1024
#define EPSV 1e-6f
#define NROWS (KD*BD*SD)      /* 8192 */
#define BS (BD*SD)            /* 2048 */
#define BSD ((long)BD*SD*DD)  /* 2097152 */
#define KBSD ((long)KD*BSD)

typedef __attribute__((ext_vector_type(2))) float v2f;
typedef __attribute__((ext_vector_type(8))) float v8f;

__global__ void mhc_init(float* acc) {
    if (threadIdx.x < 24) acc[threadIdx.x] = 0.f;
}

// One block per (n,b,s) row. Computes RMS-normed row in LDS, then streams the
// alpha tensors once, accumulating h_res (16, via WMMA f32 16x16x4 with the C
// tile as running accumulator) and h_pre/h_post (8, via VALU lanes).
__global__ __launch_bounds__(256)
void mhc_reduce(const float* __restrict__ x,
                const float* __restrict__ w,
                const float* __restrict__ res_a,
                const float* __restrict__ pre_a,
                const float* __restrict__ post_a,
                float* __restrict__ acc) {
    __shared__ float xn[DD];
    __shared__ float red[256];
    __shared__ float res_part[8*16];
    __shared__ float pp_part[8*8];

    const int t = threadIdx.x;
    const int row = blockIdx.x;
    const long rowD = (long)row * DD;

    // ---- Phase A: row sum-of-squares + normalized row into LDS ----
    float4 xv = *(const float4*)(x + rowD + 4*t);
    red[t] = xv.x*xv.x + xv.y*xv.y + xv.z*xv.z + xv.w*xv.w;
    __syncthreads();
    for (int off = 128; off > 0; off >>= 1) {
        if (t < off) red[t] += red[t + off];
        __syncthreads();
    }
    const float scale = rsqrtf(red[0] / (float)DD + EPSV);
    float4 wv = *(const float4*)(w + 4*t);
    xn[4*t+0] = xv.x * scale * wv.x;
    xn[4*t+1] = xv.y * scale * wv.y;
    xn[4*t+2] = xv.z * scale * wv.z;
    xn[4*t+3] = xv.w * scale * wv.w;
    __syncthreads();

    // ---- Phase B: streamed reduction, 4 elements per WMMA per wave ----
    const int wave = t >> 5;
    const int lane = t & 31;
    const long resBase = rowD * 16;
    const long ppBase  = rowD * 4;
    const int hi = (lane >= 16) ? 1 : 0;   // upper half-wave handles K=2,3
    const int ml = lane & 15;              // A-matrix row index (= output m)
    const int elem = lane >> 3;            // 0..3: which d element (pre/post)
    const int j = lane & 7;                // 0..7: pre(0..3)/post(4..7) column
    const float* ppmat = (j < 4) ? pre_a : post_a;
    const int jj = j & 3;

    v8f c = {};        // C/D tile = running accumulator for h_res
    float pp = 0.f;

    for (int iter = 0; iter < 32; ++iter) {
        const int d0 = (iter*8 + wave) * 4;
        // A[m][k] = res_alpha[d0+k][m]; m stride-1 in memory -> contiguous loads
        const long ao = resBase + (long)d0*16 + ml + hi*32;
        v2f a, b;
        a.x = res_a[ao];         // K = 0 (lanes<16) / K = 2 (lanes>=16)
        a.y = res_a[ao + 16];    // K = 1           / K = 3
        // B[k][n] = xn[d0+k], broadcast across N (VGPR0: rows 0,2; VGPR1: 1,3)
        b.x = xn[d0 + 2*hi];
        b.y = xn[d0 + 2*hi + 1];
        c = __builtin_amdgcn_wmma_f32_16x16x4_f32(false, a, false, b,
                                                  (short)0, c, false, false);
        // pre/post: 4 elems x 8 outputs spread over the 32 lanes
        const float xe = xn[d0 + elem];
        pp += xe * ppmat[ppBase + (long)(d0 + elem)*4 + jj];
    }

    // reduce pre/post partials: lanes {j, j+8, j+16, j+24} share output j
    pp += __shfl_xor(pp, 8);
    pp += __shfl_xor(pp, 16);
    if (lane < 8) pp_part[wave*8 + lane] = pp;
    // C column 0 holds h_res partials: (M=p, N=0) on lane 0, (M=p+8) on lane 16
    if (lane == 0) {
        #pragma unroll
        for (int p = 0; p < 8; ++p) res_part[wave*16 + p] = c[p];
    }
    if (lane == 16) {
        #pragma unroll
        for (int p = 0; p < 8; ++p) res_part[wave*16 + 8 + p] = c[p];
    }
    __syncthreads();
    if (t < 16) {
        float s = 0.f;
        for (int wv2 = 0; wv2 < 8; ++wv2) s += res_part[wv2*16 + t];
        atomicAdd(&acc[t], s);
    } else if (t < 24) {
        const int q = t - 16;
        float s = 0.f;
        for (int wv2 = 0; wv2 < 8; ++wv2) s += pp_part[wv2*8 + q];
        atomicAdd(&acc[16 + q], s);
    }
}

// Single-thread: sigmoids + 4x4 log-domain Sinkhorn (20 iters, scaling 0.05)
__global__ void mhc_maps(const float* __restrict__ acc,
                         const float* __restrict__ res_beta,
                         const float* __restrict__ pre_beta,
                         const float* __restrict__ post_beta,
                         float* __restrict__ maps) {
    if (threadIdx.x != 0 || blockIdx.x != 0) return;
    for (int k = 0; k < 4; ++k) {
        maps[16 + k] = 1.f / (1.f + expf(-(1e-4f * acc[16 + k] + pre_beta[k])));
        maps[20 + k] = 2.f / (1.f + expf(-(1e-4f * acc[20 + k] + post_beta[k])));
    }
    float Z[4][4];
    for (int i = 0; i < 4; ++i)
        for (int q = 0; q < 4; ++q)
            Z[i][q] = (1e-4f * acc[i*4 + q] + res_beta[q]) * 20.0f; // /0.05
    float u[4] = {0,0,0,0}, v[4] = {0,0,0,0};
    for (int it = 0; it < 20; ++it) {
        for (int i = 0; i < 4; ++i) {
            float m = -1e30f;
            for (int q = 0; q < 4; ++q) m = fmaxf(m, Z[i][q] + v[q]);
            float s = 0.f;
            for (int q = 0; q < 4; ++q) s += expf(Z[i][q] + v[q] - m);
            u[i] = -(m + logf(s));
        }
        for (int q = 0; q < 4; ++q) {
            float m = -1e30f;
            for (int i = 0; i < 4; ++i) m = fmaxf(m, Z[i][q] + u[i]);
            float s = 0.f;
            for (int i = 0; i < 4; ++i) s += expf(Z[i][q] + u[i] - m);
            v[q] = -(m + logf(s));
        }
    }
    for (int i = 0; i < 4; ++i)
        for (int q = 0; q < 4; ++q)
            maps[i*4 + q] = expf(Z[i][q] + u[i] + v[q]);
}

// One block per (b,s): recompute the 4 stream norms, then emit res_out and
// post_out with fully coalesced float4 stores.
__global__ __launch_bounds__(256)
void mhc_output(const float* __restrict__ x,
                const float* __restrict__ w,
                const float* __restrict__ maps,
                float* __restrict__ out) {
    __shared__ float  lx[KD][DD];
    __shared__ float4 red4[256];
    const int t = threadIdx.x;
    const int bs = blockIdx.x;

    float ssq[4];
    #pragma unroll
    for (int n = 0; n < 4; ++n) {
        float4 xv = *(const float4*)(x + ((long)n*BS + bs)*DD + 4*t);
        *(float4*)&lx[n][4*t] = xv;
        ssq[n] = xv.x*xv.x + xv.y*xv.y + xv.z*xv.z + xv.w*xv.w;
    }
    red4[t] = make_float4(ssq[0], ssq[1], ssq[2], ssq[3]);
    __syncthreads();
    for (int off = 128; off > 0; off >>= 1) {
        if (t < off) {
            float4 a = red4[t], b = red4[t + off];
            red4[t] = make_float4(a.x+b.x, a.y+b.y, a.z+b.z, a.w+b.w);
        }
        __syncthreads();
    }
    const float4 tot = red4[0];
    const float sc[4] = { rsqrtf(tot.x/(float)DD + EPSV),
                          rsqrtf(tot.y/(float)DD + EPSV),
                          rsqrtf(tot.z/(float)DD + EPSV),
                          rsqrtf(tot.w/(float)DD + EPSV) };
    float P[4][4], prem[4], postm[4];
    #pragma unroll
    for (int n = 0; n < 4; ++n) {
        for (int m = 0; m < 4; ++m) P[n][m] = maps[n*4 + m];
        prem[n]  = maps[16 + n];
        postm[n] = maps[20 + n];
    }
    float4 wv = *(const float4*)(w + 4*t);
    const float wj[4] = {wv.x, wv.y, wv.z, wv.w};
    float xnv[4][4];
    #pragma unroll
    for (int n = 0; n < 4; ++n)
        for (int q = 0; q < 4; ++q)
            xnv[n][q] = lx[n][4*t + q] * sc[n] * wj[q];

    #pragma unroll
    for (int m = 0; m < 4; ++m) {
        float o[4];
        for (int q = 0; q < 4; ++q) {
            float s = 0.f;
            for (int n = 0; n < 4; ++n) s += xnv[n][q] * P[n][m];
            o[q] = s;
        }
        *(float4*)(out + ((long)m*BS + bs)*DD + 4*t) =
            make_float4(o[0], o[1], o[2], o[3]);
    }
    float li[4];
    #pragma unroll
    for (int q = 0; q < 4; ++q) {
        float s = 0.f;
        for (int n = 0; n < 4; ++n) s += xnv[n][q] * prem[n];
        li[q] = s;
    }
    #pragma unroll
    for (int n = 0; n < 4; ++n) {
        *(float4*)(out + KBSD + ((long)n*BS + bs)*DD + 4*t) =
            make_float4(li[0]*postm[n], li[1]*postm[n], li[2]*postm[n], li[3]*postm[n]);
    }
}

extern "C" void kernel_launch(void* const* d_in, const int* in_sizes, int n_in,
                              void* d_out, int out_size, void* d_ws, size_t ws_size,
                              hipStream_t stream) {
    const float* x      = (const float*)d_in[0];
    const float* w      = (const float*)d_in[1];
    const float* res_a  = (const float*)d_in[2];
    const float* pre_a  = (const float*)d_in[3];
    const float* post_a = (const float*)d_in[4];
    const float* res_b  = (const float*)d_in[5];
    const float* pre_b  = (const float*)d_in[6];
    const float* post_b = (const float*)d_in[7];
    float* acc  = (float*)d_ws;        // [0..23] h accumulators
    float* maps = acc + 32;            // [0..15] res_map, [16..19] pre, [20..23] post
    float* out  = (float*)d_out;

    mhc_init  <<<1, 32, 0, stream>>>(acc);
    mhc_reduce<<<NROWS, 256, 0, stream>>>(x, w, res_a, pre_a, post_a, acc);
    mhc_maps  <<<1, 32, 0, stream>>>(acc, res_b, pre_b, post_b, maps);
    mhc_output<<<BS, 256, 0, stream>>>(x, w, maps, out);
}